// VoxelGraph_4870492914283
// MI455X (gfx1250) — compile-verified
//
#include <hip/hip_runtime.h>
#include <math.h>

typedef __attribute__((ext_vector_type(2))) float v2f;
typedef __attribute__((ext_vector_type(8))) float v8f;

#define WAVES_PER_BLOCK 8
#define BLOCK_THREADS   (WAVES_PER_BLOCK * 32)
#define Q_PER_WAVE      32
#define Q_PER_BLOCK     (WAVES_PER_BLOCK * Q_PER_WAVE)   // 256
#define REF_CHUNK       64                                // refs staged per block iteration
#define SUB             16                                // refs per WMMA tile
#define K_NEI           32

__device__ __forceinline__ int iabs_(int v) { return v < 0 ? -v : v; }

__global__ __launch_bounds__(BLOCK_THREADS)
void voxel_graph_wmma_kernel(const float* __restrict__ ref,
                             const float* __restrict__ query,
                             int nRef, int nQuery,
                             int* __restrict__ outEdges,    // [2, nQuery*K_NEI] int32
                             float* __restrict__ outDist)   // [nQuery, K_NEI]
{
    __shared__ float4 sRef[REF_CHUNK];                    // (rx, ry, rz, |r|^2)
    __shared__ int4   sVox[REF_CHUNK];                    // (vb, vx, vy, vz)
    __shared__ float4 sQ[Q_PER_BLOCK];                    // (qx, qy, qz, |q|^2)
    __shared__ float  sD2[WAVES_PER_BLOCK][Q_PER_WAVE][SUB];

    const int tid  = threadIdx.x;
    const int wave = tid >> 5;
    const int lane = tid & 31;
    const int qbase = blockIdx.x * Q_PER_BLOCK + wave * Q_PER_WAVE;
    const int q = qbase + lane;

    // ---- load this lane's query, voxelize exactly like the reference ----
    float4 qp = make_float4(0.0f, 0.0f, 0.0f, 0.0f);
    if (q < nQuery) qp = reinterpret_cast<const float4*>(query)[q];  // (b,x,y,z)
    const float qx = qp.y, qy = qp.z, qz = qp.w;
    const int qvb = (int)floorf((qp.x - 0.0f)  / 0.999f);
    const int qvx = (int)floorf((qx   - 0.0f)  / 0.4f);
    const int qvy = (int)floorf((qy   + 40.0f) / 0.4f);
    const int qvz = (int)floorf((qz   + 3.0f)  / 0.4f);
    const float q2 = qx*qx + qy*qy + qz*qz;
    sQ[wave * 32 + lane] = make_float4(qx, qy, qz, q2);
    __syncthreads();

    // ---- build WMMA A operands (16x4 f32: lanes 0-15 hold K=0,1; lanes 16-31 K=2,3) ----
    const int half = lane >> 4;       // K-half selector
    const int m    = lane & 15;       // matrix row (A) / column (B)
    const float4 qa0 = sQ[wave * 32 + m];
    const float4 qa1 = sQ[wave * 32 + 16 + m];
    v2f a0, a1;
    a0.x = half ? (-2.0f * qa0.z) : (-2.0f * qa0.x);   // K=2 -> -2qz ; K=0 -> -2qx
    a0.y = half ? 1.0f            : (-2.0f * qa0.y);   // K=3 -> 1    ; K=1 -> -2qy
    a1.x = half ? (-2.0f * qa1.z) : (-2.0f * qa1.x);
    a1.y = half ? 1.0f            : (-2.0f * qa1.y);

    // ---- C operands: |q|^2 broadcast along rows (C/D layout: vgpr v -> M = v + 8*half) ----
    v8f c0, c1;
#pragma unroll
    for (int v = 0; v < 8; ++v) {
        c0[v] = sQ[wave * 32 + (v + 8 * half)].w;
        c1[v] = sQ[wave * 32 + 16 + (v + 8 * half)].w;
    }

    // ---- register-resident stable top-32 (ascending d2, ties -> earlier ref idx) ----
    float bd[K_NEI];
    int   bi[K_NEI];
#pragma unroll
    for (int s = 0; s < K_NEI; ++s) { bd[s] = __builtin_inff(); bi[s] = -1; }

    const float4* ref4 = reinterpret_cast<const float4*>(ref);

    for (int r0 = 0; r0 < nRef; r0 += REF_CHUNK) {
        // ---- stage a chunk of refs into LDS (coords + |r|^2 + voxel ints) ----
        if (tid < REF_CHUNK) {
            const int ri = r0 + tid;
            float4 rp = make_float4(0.0f, 0.0f, 0.0f, 0.0f);
            if (ri < nRef) rp = ref4[ri];
            const float rx = rp.y, ry = rp.z, rz = rp.w;
            int4 vox;
            if (ri < nRef) {
                vox.x = (int)floorf((rp.x - 0.0f)  / 0.999f);
                vox.y = (int)floorf((rx   - 0.0f)  / 0.4f);
                vox.z = (int)floorf((ry   + 40.0f) / 0.4f);
                vox.w = (int)floorf((rz   + 3.0f)  / 0.4f);
            } else {
                vox = make_int4(0x40000000, 0x40000000, 0x40000000, 0x40000000);
            }
            sRef[tid] = make_float4(rx, ry, rz, rx*rx + ry*ry + rz*rz);
            sVox[tid] = vox;
            if (ri + REF_CHUNK < nRef)
                __builtin_prefetch(&ref4[ri + REF_CHUNK], 0, 0);   // global_prefetch_b8
        }
        __syncthreads();

#pragma unroll
        for (int sc = 0; sc < REF_CHUNK / SUB; ++sc) {
            const int jb = sc * SUB;

            // B operand (4x16 f32): lanes 0-15 hold K=0,1 of col n; lanes 16-31 K=2,3
            const float4 rc = sRef[jb + m];
            v2f bm;
            bm.x = half ? rc.z : rc.x;   // K=2 -> rz   ; K=0 -> rx
            bm.y = half ? rc.w : rc.y;   // K=3 -> |r|^2; K=1 -> ry

            // d2 tile = A(16x4) x B(4x16) + C  -> full |q|^2 + |r|^2 - 2 q.r
            v8f d0 = __builtin_amdgcn_wmma_f32_16x16x4_f32(
                false, a0, false, bm, (short)0, c0, false, false);
            v8f d1 = __builtin_amdgcn_wmma_f32_16x16x4_f32(
                false, a1, false, bm, (short)0, c1, false, false);

            // transpose d2 tiles into [query][ref] layout via LDS
#pragma unroll
            for (int v = 0; v < 8; ++v) {
                sD2[wave][v + 8 * half][m]      = d0[v];
                sD2[wave][16 + v + 8 * half][m] = d1[v];
            }
            __syncthreads();

            // each lane owns one query: scan 16 refs, mask, insert into sorted top-32
            for (int j = 0; j < SUB; ++j) {
                const int4 rv = sVox[jb + j];
                const bool ok = (rv.x == qvb) &
                                (iabs_(rv.y - qvx) <= 1) &
                                (iabs_(rv.z - qvy) <= 1) &
                                (iabs_(rv.w - qvz) <= 1);
                if (ok) {
                    const float d = sD2[wave][lane][j];
                    if (d < bd[K_NEI - 1]) {
                        const int ridx = r0 + jb + j;
#pragma unroll
                        for (int s = K_NEI - 1; s >= 0; --s) {
                            const float prev  = (s > 0) ? bd[s - 1] : -__builtin_inff();
                            const int   previ = (s > 0) ? bi[s - 1] : -1;
                            if (d < prev)            { bd[s] = prev; bi[s] = previ; }
                            else if (d < bd[s])      { bd[s] = d;    bi[s] = ridx;  }
                        }
                    }
                }
            }
            __syncthreads();   // reads done before next tile overwrites sD2 / sRef
        }
    }

    // ---- emit edges (int32, -1 padded) and dists (f32, 0 padded) ----
    if (q < nQuery) {
        const int MK = nQuery * K_NEI;
#pragma unroll
        for (int s = 0; s < K_NEI; ++s) {
            const bool valid = bd[s] < __builtin_inff();
            outEdges[q * K_NEI + s]      = valid ? bi[s] : -1;
            outEdges[MK + q * K_NEI + s] = valid ? q     : -1;
            outDist[q * K_NEI + s]       = valid ? bd[s] : 0.0f;
        }
    }
}

extern "C" void kernel_launch(void* const* d_in, const int* in_sizes, int n_in,
                              void* d_out, int out_size, void* d_ws, size_t ws_size,
                              hipStream_t stream) {
    (void)n_in; (void)d_ws; (void)ws_size; (void)out_size;
    const float* ref   = (const float*)d_in[0];   // [nRef, 4]  (b,x,y,z)
    const float* query = (const float*)d_in[1];   // [nQuery, 4]
    const int nRef   = in_sizes[0] / 4;
    const int nQuery = in_sizes[1] / 4;

    int*   outEdges = (int*)d_out;                               // [2, nQuery*32]
    float* outDist  = (float*)d_out + (size_t)2 * nQuery * K_NEI; // [nQuery, 32]

    const int blocks = (nQuery + Q_PER_BLOCK - 1) / Q_PER_BLOCK;
    hipLaunchKernelGGL(voxel_graph_wmma_kernel, dim3(blocks), dim3(BLOCK_THREADS),
                       0, stream, ref, query, nRef, nQuery, outEdges, outDist);
}